// PolicyNetwork_20117626814588
// MI455X (gfx1250) — compile-verified
//
#include <hip/hip_runtime.h>

// Problem constants (from the reference)
#define RR 512        // robots
#define TT 512        // tasks
#define DD 256        // per-entity embedding dim
#define HH 512        // hidden dim (== INPUT_DIM)

typedef float v2f __attribute__((ext_vector_type(2)));
typedef float v8f __attribute__((ext_vector_type(8)));

// -----------------------------------------------------------------------------
// Stage 1: U[0:512,:]    = robots @ W1[:256,:]               (rpart)
//          U[512:1024,:] = tasks  @ W1[256:,:] + b1          (tpart + bias)
// One wave32 per 16x64 output tile: 4 f32 accumulators share one A fragment
// per K-step -> 4x A-traffic reuse, 256 v_wmma_f32_16x16x4_f32 per wave.
// grid = (HH/64, (RR+TT)/16) = (8, 64), block = 32 (one wave).
// -----------------------------------------------------------------------------
__global__ __launch_bounds__(32)
void policy_stage1_wmma(const float* __restrict__ robots,
                        const float* __restrict__ tasks,
                        const float* __restrict__ W1,
                        const float* __restrict__ b1,
                        float* __restrict__ U)
{
    const int lane = threadIdx.x;        // 0..31
    const int half = lane >> 4;          // 0: K=k,k+1   1: K=k+2,k+3
    const int l    = lane & 15;

    const int n0      = blockIdx.x * 64 + l;   // first of 4 column tiles
    const int rowTile = blockIdx.y * 16;       // 0..1023, step 16
    const bool isTask = (rowTile >= RR);

    const float* __restrict__ X = isTask ? tasks : robots;
    const int xrow = (isTask ? rowTile - RR : rowTile) + l;  // A row for this lane
    const int kOff = isTask ? DD : 0;                        // W1 row offset

    v8f c0 = {}, c1 = {}, c2 = {}, c3 = {};

    for (int k = 0; k < DD; k += 4) {
        const int ka = k + 2 * half;
        // A fragment: 16x4 f32 (two consecutive K per lane, 8B-aligned) — shared by 4 WMMAs
        v2f a = *(const v2f*)(X + xrow * DD + ka);
        // B fragments: 4x16 f32 per column tile, col = n0 + 16j, rows ka/ka+1 per lane
        const float* __restrict__ w = W1 + (kOff + ka) * HH + n0;
        v2f b0 = { w[ 0], w[HH +  0] };
        v2f b1v = { w[16], w[HH + 16] };
        v2f b2v = { w[32], w[HH + 32] };
        v2f b3v = { w[48], w[HH + 48] };
        c0 = __builtin_amdgcn_wmma_f32_16x16x4_f32(false, a, false, b0,  (short)0, c0, false, false);
        c1 = __builtin_amdgcn_wmma_f32_16x16x4_f32(false, a, false, b1v, (short)0, c1, false, false);
        c2 = __builtin_amdgcn_wmma_f32_16x16x4_f32(false, a, false, b2v, (short)0, c2, false, false);
        c3 = __builtin_amdgcn_wmma_f32_16x16x4_f32(false, a, false, b3v, (short)0, c3, false, false);
    }

    // D layout: VGPR i -> M = i (lanes 0-15) or 8+i (lanes 16-31), N = lane&15
    const int rowBase = rowTile + 8 * half;
    #pragma unroll
    for (int j = 0; j < 4; ++j) {
        const v8f c = (j == 0) ? c0 : (j == 1) ? c1 : (j == 2) ? c2 : c3;
        const int n = n0 + 16 * j;
        const float bias = isTask ? b1[n] : 0.0f;   // fold b1 into the task half
        #pragma unroll
        for (int i = 0; i < 8; ++i) {
            U[(size_t)(rowBase + i) * HH + n] = c[i] + bias;
        }
    }
}

// -----------------------------------------------------------------------------
// Stage 2: scores[r,t] = sum_h relu(rpart[r,h] + tpartb[t,h]) * W2[h] + b2
// 32x32 (r,t) tile per 256-thread block. h is pipelined through LDS in chunks
// of 64 using ASYNC loads (global_load_async_to_lds_b32, ASYNCcnt-tracked)
// with double buffering: compute chunk c while the async units fill chunk c+1.
// Tiles are stored transposed [h][r] (stride 33) so compute reads are
// bank-conflict-free (lane-varying r -> 32 distinct banks).
// grid = (RR/32, TT/32) = (16,16), block = 256 (8 waves).
// -----------------------------------------------------------------------------
#define S2_TR 32
#define S2_TT 32
#define S2_HC 64
#define S2_PAD (S2_TR + 1)   // 33 floats, conflict-free lane-varying reads

__device__ __forceinline__ void async_tile_load_T(float* __restrict__ dstT, // [S2_HC][S2_PAD]
                                                  const float* __restrict__ src,
                                                  int row0, int h0, int tid)
{
    // 32 rows x 64 h, transposed into LDS. Consecutive tid -> consecutive h
    // (global-coalesced). Each iteration is one wave-wide async b32 (128B/wave).
    #pragma unroll
    for (int k = 0; k < (S2_TR * S2_HC) / 256; ++k) {
        const int i  = tid + k * 256;
        const int rr = i >> 6;               // row in tile
        const int hh = i & (S2_HC - 1);
        const unsigned int lds = (unsigned int)(uintptr_t)(dstT + hh * S2_PAD + rr);
        const unsigned long long g =
            (unsigned long long)(uintptr_t)(src + (size_t)(row0 + rr) * HH + h0 + hh);
        asm volatile("global_load_async_to_lds_b32 %0, %1, off"
                     :: "v"(lds), "v"(g) : "memory");
    }
}

__global__ __launch_bounds__(256)
void policy_stage2_score(const float* __restrict__ U,
                         const float* __restrict__ W2,
                         const float* __restrict__ b2,
                         float* __restrict__ out)
{
    __shared__ float rpT[2][S2_HC][S2_PAD];  // double-buffered, transposed [h][r]
    __shared__ float tpT[2][S2_HC][S2_PAD];
    __shared__ float w2s[HH];

    const float* __restrict__ rpart = U;                      // [512,512]
    const float* __restrict__ tpart = U + (size_t)RR * HH;    // [512,512] (b1 folded)

    const int tid = threadIdx.x;
    const int r0  = blockIdx.x * S2_TR;
    const int t0  = blockIdx.y * S2_TT;

    const int rl     = tid & 31;          // robot within tile (lane-varying)
    const int tlBase = (tid >> 5) * 4;    // 4 tasks per thread (wave-uniform)

    // One-time: whole W2 into LDS (2 KB).
    for (int i = tid; i < HH; i += 256) w2s[i] = W2[i];

    // Prologue: async-fill buffer 0 with chunk 0.
    async_tile_load_T(&rpT[0][0][0], rpart, r0, 0, tid);
    async_tile_load_T(&tpT[0][0][0], tpart, t0, 0, tid);

    float acc0 = 0.f, acc1 = 0.f, acc2 = 0.f, acc3 = 0.f;

    for (int c = 0; c < HH / S2_HC; ++c) {
        const int buf = c & 1;

        // Our async writes to LDS are done...
        asm volatile("s_wait_asynccnt 0x0" ::: "memory");
        // ...and visible to every wave in the block.
        __syncthreads();

        // Kick off the next chunk into the other buffer. Safe: the barrier
        // above proves all waves finished computing on that buffer (chunk c-1).
        if (c + 1 < HH / S2_HC) {
            const int h1 = (c + 1) * S2_HC;
            async_tile_load_T(&rpT[buf ^ 1][0][0], rpart, r0, h1, tid);
            async_tile_load_T(&tpT[buf ^ 1][0][0], tpart, t0, h1, tid);
        }

        const int h0 = c * S2_HC;
        #pragma unroll 4
        for (int hh = 0; hh < S2_HC; ++hh) {
            const float w  = w2s[h0 + hh];        // broadcast
            const float rv = rpT[buf][hh][rl];    // conflict-free
            const float t0v = tpT[buf][hh][tlBase + 0];  // wave-uniform -> broadcast
            const float t1v = tpT[buf][hh][tlBase + 1];
            const float t2v = tpT[buf][hh][tlBase + 2];
            const float t3v = tpT[buf][hh][tlBase + 3];
            acc0 += fmaxf(rv + t0v, 0.f) * w;
            acc1 += fmaxf(rv + t1v, 0.f) * w;
            acc2 += fmaxf(rv + t2v, 0.f) * w;
            acc3 += fmaxf(rv + t3v, 0.f) * w;
        }
        __syncthreads();   // all waves done with buf before it is refilled next round
    }

    const float bb = b2[0];
    float* o = out + (size_t)(r0 + rl) * TT + t0 + tlBase;
    o[0] = acc0 + bb;
    o[1] = acc1 + bb;
    o[2] = acc2 + bb;
    o[3] = acc3 + bb;
}

// -----------------------------------------------------------------------------
// Launcher
// inputs: 0=robot_embeddings [512,256], 1=task_embeddings [512,256],
//         2=W1 [512,512], 3=b1 [512], 4=W2 [512,1], 5=b2 [1]
// output: scores [512,512] f32
// workspace: U = [rpart(512x512) ; tpartb(512x512)] = 2 MB f32
// -----------------------------------------------------------------------------
extern "C" void kernel_launch(void* const* d_in, const int* in_sizes, int n_in,
                              void* d_out, int out_size, void* d_ws, size_t ws_size,
                              hipStream_t stream) {
    const float* robots = (const float*)d_in[0];
    const float* tasks  = (const float*)d_in[1];
    const float* W1     = (const float*)d_in[2];
    const float* b1     = (const float*)d_in[3];
    const float* W2     = (const float*)d_in[4];
    const float* b2     = (const float*)d_in[5];

    float* U   = (float*)d_ws;          // 1024*512 floats = 2 MB
    float* out = (float*)d_out;

    dim3 g1(HH / 64, (RR + TT) / 16);   // (8, 64)
    policy_stage1_wmma<<<g1, 32, 0, stream>>>(robots, tasks, W1, b1, U);

    dim3 g2(RR / S2_TR, TT / S2_TT);    // (16, 16)
    policy_stage2_score<<<g2, 256, 0, stream>>>(U, W2, b2, out);
}